// AttentionModule_58729382805941
// MI455X (gfx1250) — compile-verified
//
#include <hip/hip_runtime.h>
#include <math.h>

typedef __attribute__((ext_vector_type(2))) float v2f;
typedef __attribute__((ext_vector_type(8))) float v8f;

#define DIMX 64
#define HID 128
#define OUTD 16
#define BATCH 32
#define NC 4
#define VCH 5
#define IMG 224
#define PNUM (IMG * IMG)

// ---------------------------------------------------------------------------
// MLP + LayerNorm head, fp32 WMMA (V_WMMA_F32_16X16X4_F32), one wave32 block
// per 16-row tile.
//   mode 0: X row r, col d = in[r*64 + d]                  (q: (B,64,1))
//   mode 1: X row r=(b*4+c), col d = in[b*256 + d*4 + c]   (k: (B,64,1,4))
// out16: row-major (rows x 16), LayerNorm'd.
// ---------------------------------------------------------------------------
__global__ __launch_bounds__(32) void mlp_ln_kernel(
    const float* __restrict__ X, const float* __restrict__ W1,
    const float* __restrict__ b1, const float* __restrict__ W2,
    const float* __restrict__ b2, const float* __restrict__ g,
    const float* __restrict__ beta, float* __restrict__ out16, int mode) {
  __shared__ float Hs[16][HID + 1];  // +1 pad: conflict-free column reads
  __shared__ float h2[16][OUTD + 1];

  const int lane = threadIdx.x;   // 0..31, wave32
  const int rl = lane & 15;       // row/col-in-half
  const int half = lane >> 4;     // 0|1
  const int m_base = blockIdx.x * 16;

  // Per-lane A-operand addressing into X for this tile's rows.
  long abase;
  int astride;
  if (mode == 0) {
    abase = (long)(m_base + rl) * DIMX;
    astride = 1;
  } else {
    int r = m_base + rl;
    abase = (long)(r >> 2) * (DIMX * NC) + (r & 3);
    astride = NC;
  }

  // ---- Layer 1: H = leakyrelu(X @ W1 + b1), tile 16 x 128 ----
  for (int nt = 0; nt < HID / 16; ++nt) {
    v8f acc = {};
    for (int kk = 0; kk < DIMX / 4; ++kk) {
      const int k0 = kk * 4 + 2 * half;
      v2f a, b;
      a.x = X[abase + (long)(k0 + 0) * astride];
      a.y = X[abase + (long)(k0 + 1) * astride];
      b.x = W1[(k0 + 0) * HID + nt * 16 + rl];
      b.y = W1[(k0 + 1) * HID + nt * 16 + rl];
      acc = __builtin_amdgcn_wmma_f32_16x16x4_f32(false, a, false, b,
                                                  (short)0, acc, false, false);
    }
    const float bias = b1[nt * 16 + rl];
#pragma unroll
    for (int r = 0; r < 8; ++r) {
      float h = acc[r] + bias;
      h = (h > 0.0f) ? h : 0.1f * h;       // LeakyReLU(0.1)
      Hs[r + 8 * half][nt * 16 + rl] = h;  // row = D's M, col = D's N
    }
  }
  __syncthreads();

  // ---- Layer 2: h2 = H @ W2 + b2, tile 16 x 16 ----
  v8f acc2 = {};
  for (int kk = 0; kk < HID / 4; ++kk) {
    const int k0 = kk * 4 + 2 * half;
    v2f a, b;
    a.x = Hs[rl][k0 + 0];
    a.y = Hs[rl][k0 + 1];
    b.x = W2[(k0 + 0) * OUTD + rl];
    b.y = W2[(k0 + 1) * OUTD + rl];
    acc2 = __builtin_amdgcn_wmma_f32_16x16x4_f32(false, a, false, b, (short)0,
                                                 acc2, false, false);
  }
  const float bias2 = b2[rl];
#pragma unroll
  for (int r = 0; r < 8; ++r) h2[r + 8 * half][rl] = acc2[r] + bias2;
  __syncthreads();

  // ---- LayerNorm(16) per row, lanes 0..15 each own one row ----
  if (lane < 16) {
    float mu = 0.0f;
#pragma unroll
    for (int j = 0; j < OUTD; ++j) mu += h2[lane][j];
    mu *= (1.0f / OUTD);
    float var = 0.0f;
#pragma unroll
    for (int j = 0; j < OUTD; ++j) {
      const float d = h2[lane][j] - mu;
      var += d * d;
    }
    var *= (1.0f / OUTD);
    const float rs = rsqrtf(var + 1e-5f);
#pragma unroll
    for (int j = 0; j < OUTD; ++j)
      out16[(m_base + lane) * OUTD + j] = (h2[lane][j] - mu) * rs * g[j] + beta[j];
  }
}

// ---------------------------------------------------------------------------
// logits -> modality dropout -> softmax(T=10). One wave, lane b handles batch b.
// ---------------------------------------------------------------------------
__global__ __launch_bounds__(32) void logits_kernel(
    const float* __restrict__ q16, const float* __restrict__ k16,
    const int* __restrict__ md, float* __restrict__ att) {
  const int b = threadIdx.x;  // 0..31 == BATCH
  const float scale = 0.125f; // DIM^-0.5 = 1/8
  float lg[NC];
#pragma unroll
  for (int c = 0; c < NC; ++c) {
    float s = 0.0f;
#pragma unroll
    for (int j = 0; j < OUTD; ++j)
      s += q16[b * OUTD + j] * k16[(b * NC + c) * OUTD + j];
    lg[c] = (s * scale - (float)md[b * NC + c] * 1.0e5f) * 0.1f;  // /TEMP
  }
  float mx = fmaxf(fmaxf(lg[0], lg[1]), fmaxf(lg[2], lg[3]));
  float e[NC], se = 0.0f;
#pragma unroll
  for (int c = 0; c < NC; ++c) {
    e[c] = __expf(lg[c] - mx);
    se += e[c];
  }
  const float inv = 1.0f / se;
#pragma unroll
  for (int c = 0; c < NC; ++c) att[b * NC + c] = e[c] * inv;
}

// ---------------------------------------------------------------------------
// Bandwidth-bound fuse pass: one thread per (batch, pixel).
// mask (B,1,IMG,IMG,C) and v (B,5,P,C) are C-contiguous -> float4 loads.
// ---------------------------------------------------------------------------
__global__ __launch_bounds__(256) void fuse_kernel(
    const float* __restrict__ mask, const float* __restrict__ v,
    const float* __restrict__ att, float* __restrict__ fused,
    float* __restrict__ attn_out) {
  const long idx = (long)blockIdx.x * blockDim.x + threadIdx.x;  // b*P + p
  if (idx >= (long)BATCH * PNUM) return;
  const int b = (int)(idx / PNUM);
  const long p = idx - (long)b * PNUM;

  const float4 m4 = reinterpret_cast<const float4*>(mask)[idx];
  const float4 a4 = reinterpret_cast<const float4*>(att)[b];

  float w0 = a4.x * m4.x, w1 = a4.y * m4.y, w2 = a4.z * m4.z, w3 = a4.w * m4.w;
  const float inv = 1.0f / (w0 + w1 + w2 + w3 + 1e-8f);
  w0 *= inv; w1 *= inv; w2 *= inv; w3 *= inv;

  const float4* vb = reinterpret_cast<const float4*>(v) + (long)b * VCH * PNUM + p;
  const long fb = (long)b * VCH * PNUM + p;
#pragma unroll
  for (int ch = 0; ch < VCH; ++ch) {
    const float4 vv = vb[(long)ch * PNUM];
    fused[fb + (long)ch * PNUM] = w0 * vv.x + w1 * vv.y + w2 * vv.z + w3 * vv.w;
  }
  const long ab = (long)b * NC * PNUM + p;
  attn_out[ab] = w0;
  attn_out[ab + PNUM] = w1;
  attn_out[ab + 2L * PNUM] = w2;
  attn_out[ab + 3L * PNUM] = w3;
}

extern "C" void kernel_launch(void* const* d_in, const int* in_sizes, int n_in,
                              void* d_out, int out_size, void* d_ws,
                              size_t ws_size, hipStream_t stream) {
  const float* q    = (const float*)d_in[0];
  const float* k    = (const float*)d_in[1];
  const float* v    = (const float*)d_in[2];
  const float* mask = (const float*)d_in[3];
  const int*   md   = (const int*)d_in[4];
  const float* Wq1 = (const float*)d_in[5];
  const float* bq1 = (const float*)d_in[6];
  const float* Wq2 = (const float*)d_in[7];
  const float* bq2 = (const float*)d_in[8];
  const float* gq    = (const float*)d_in[9];
  const float* betaq = (const float*)d_in[10];
  const float* Wk1 = (const float*)d_in[11];
  const float* bk1 = (const float*)d_in[12];
  const float* Wk2 = (const float*)d_in[13];
  const float* bk2 = (const float*)d_in[14];
  const float* gk    = (const float*)d_in[15];
  const float* betak = (const float*)d_in[16];

  float* ws = (float*)d_ws;
  float* q16 = ws;                 // 32*16
  float* k16 = q16 + BATCH * OUTD; // 128*16
  float* att = k16 + BATCH * NC * OUTD;  // 32*4

  float* fused = (float*)d_out;                          // (B,5,IMG,IMG)
  float* attn  = fused + (long)BATCH * VCH * PNUM;       // (B,4,IMG,IMG)

  mlp_ln_kernel<<<BATCH / 16, 32, 0, stream>>>(q, Wq1, bq1, Wq2, bq2, gq,
                                               betaq, q16, 0);
  mlp_ln_kernel<<<(BATCH * NC) / 16, 32, 0, stream>>>(k, Wk1, bk1, Wk2, bk2,
                                                      gk, betak, k16, 1);
  logits_kernel<<<1, 32, 0, stream>>>(q16, k16, md, att);

  const long npix = (long)BATCH * PNUM;
  fuse_kernel<<<(int)((npix + 255) / 256), 256, 0, stream>>>(mask, v, att,
                                                             fused, attn);
}